// AdaptiveNoisingModule_4123168604387
// MI455X (gfx1250) — compile-verified
//
#include <hip/hip_runtime.h>
#include <hip/hip_bf16.h>
#include <math.h>

// Problem constants (from reference setup_inputs)
#define RB      8            // B
#define RN      1024         // N
#define RROWS   8192         // B*N
#define DD      1024         // D
#define MM      16384        // M
#define KNN     9
#define XPAD    1028         // LDS row stride in floats (bank-conflict-free b64 reads)
#define EPSV    1e-8f

typedef __attribute__((ext_vector_type(2))) float v2f;
typedef __attribute__((ext_vector_type(8))) float v8f;

// ---------------------------------------------------------------------------
// Kernel 1: row sum-of-squares (used for x2[] and m2[])
// ---------------------------------------------------------------------------
__global__ __launch_bounds__(256) void rowsq_kernel(const float* __restrict__ src,
                                                    float* __restrict__ out) {
  __shared__ float red[256];
  const int row = blockIdx.x;
  const float* p = src + (size_t)row * DD;
  float s = 0.0f;
  for (int c = threadIdx.x; c < DD; c += 256) {
    float v = p[c];
    s += v * v;
  }
  red[threadIdx.x] = s;
  __syncthreads();
  for (int off = 128; off > 0; off >>= 1) {
    if (threadIdx.x < off) red[threadIdx.x] += red[threadIdx.x + off];
    __syncthreads();
  }
  if (threadIdx.x == 0) out[row] = red[0];
}

// ---------------------------------------------------------------------------
// Kernel 2: fused f32-WMMA distance GEMM + on-the-fly top-9 selection.
// Block = 256 threads = 8 waves, handles 16 feature rows.
// Wave w computes C = MemTile(16x4) x FeatTile(4x16) accumulated over D,
// so C(m,n) = dot(mem_m, x_n): lane n holds 8 candidates for x-row n.
// ---------------------------------------------------------------------------
__global__ __launch_bounds__(256) void dist_topk_wmma_kernel(
    const float* __restrict__ feat,   // [RROWS][DD]
    const float* __restrict__ mem,    // [MM][DD]
    const float* __restrict__ x2,     // [RROWS]
    const float* __restrict__ m2,     // [MM]
    float* __restrict__ out_dist,     // [RROWS][KNN] (ascending)
    int*   __restrict__ nbr_idx,      // [RROWS][KNN]
    float* __restrict__ dmean)        // [RROWS]
{
  __shared__ float Xs[16 * XPAD];     // 65792 B; reused for the merge phase

  const int tid  = threadIdx.x;
  const int lane = tid & 31;
  const int wave = tid >> 5;
  const int n    = lane & 15;         // feature-row column within tile
  const int half = lane >> 4;         // 0 | 1
  const int xbase = blockIdx.x * 16;

  // Stage the 16x1024 feature tile into LDS (coalesced float4 loads).
  for (int i = tid; i < 16 * (DD / 4); i += 256) {
    const int r = i / (DD / 4);
    const int c = (i % (DD / 4)) * 4;
    const float4 v = *reinterpret_cast<const float4*>(&feat[(size_t)(xbase + r) * DD + c]);
    float* dst = &Xs[r * XPAD + c];
    dst[0] = v.x; dst[1] = v.y; dst[2] = v.z; dst[3] = v.w;
  }
  __syncthreads();

  // Per-lane sorted (ascending) top-9 of squared distances.
  float bs[KNN];
  int   bi[KNN];
#pragma unroll
  for (int t = 0; t < KNN; ++t) { bs[t] = 3.0e38f; bi[t] = 0; }

  const float x2n = x2[xbase + n];

  for (int mt = wave * 16; mt < MM; mt += 128) {
    if (mt + 128 < MM)  // hint next tile of the (L2-resident) memory bank
      __builtin_prefetch(&mem[(size_t)(mt + 128 + n) * DD], 0, 3);

    // A fragment: mem row (mt + n), K pair selected by half.
    const float* ap = &mem[(size_t)(mt + n) * DD + 2 * half];
    // B fragment: feature row n from LDS, same K addressing.
    const float* bp = &Xs[n * XPAD + 2 * half];

    v8f c0 = {0.f, 0.f, 0.f, 0.f, 0.f, 0.f, 0.f, 0.f};
    v8f c1 = {0.f, 0.f, 0.f, 0.f, 0.f, 0.f, 0.f, 0.f};
#pragma unroll 4
    for (int k0 = 0; k0 < DD; k0 += 8) {
      v2f a0 = *reinterpret_cast<const v2f*>(ap + k0);
      v2f b0 = *reinterpret_cast<const v2f*>(bp + k0);
      v2f a1 = *reinterpret_cast<const v2f*>(ap + k0 + 4);
      v2f b1 = *reinterpret_cast<const v2f*>(bp + k0 + 4);
      c0 = __builtin_amdgcn_wmma_f32_16x16x4_f32(false, a0, false, b0,
                                                 (short)0, c0, false, false);
      c1 = __builtin_amdgcn_wmma_f32_16x16x4_f32(false, a1, false, b1,
                                                 (short)0, c1, false, false);
    }
    v8f cc = c0 + c1;

    // m2 for this lane's 8 candidate mem rows.
    float m2v[8];
    const float* m2p = &m2[mt + half * 8];
    *reinterpret_cast<float4*>(&m2v[0]) = *reinterpret_cast<const float4*>(m2p);
    *reinterpret_cast<float4*>(&m2v[4]) = *reinterpret_cast<const float4*>(m2p + 4);

#pragma unroll
    for (int r = 0; r < 8; ++r) {
      const float sq  = x2n + m2v[r] - 2.0f * cc[r];
      const int   idx = mt + half * 8 + r;
      if (sq < bs[KNN - 1]) {
        bs[KNN - 1] = sq;
        bi[KNN - 1] = idx;
#pragma unroll
        for (int t = KNN - 1; t > 0; --t) {
          if (bs[t] < bs[t - 1]) {
            float fs = bs[t]; bs[t] = bs[t - 1]; bs[t - 1] = fs;
            int   fi = bi[t]; bi[t] = bi[t - 1]; bi[t - 1] = fi;
          }
        }
      }
    }
  }

  // Merge 16 contributor lists per feature row through LDS (reuse Xs space).
  __syncthreads();               // everyone done reading Xs as B-fragments
  float* ms = Xs;                // 16 rows x 144 floats = 2304 floats
  int*   mi = reinterpret_cast<int*>(&Xs[16 * 144]);
  const int contrib = wave * 2 + half;   // 0..15
#pragma unroll
  for (int t = 0; t < KNN; ++t) {
    ms[n * 144 + contrib * KNN + t] = bs[t];
    mi[n * 144 + contrib * KNN + t] = bi[t];
  }
  __syncthreads();

  if (tid < 16) {
    float fs[KNN]; int fi[KNN];
#pragma unroll
    for (int t = 0; t < KNN; ++t) { fs[t] = 3.0e38f; fi[t] = 0; }
    for (int j = 0; j < 144; ++j) {
      const float sq = ms[tid * 144 + j];
      if (sq < fs[KNN - 1]) {
        fs[KNN - 1] = sq;
        fi[KNN - 1] = mi[tid * 144 + j];
#pragma unroll
        for (int t = KNN - 1; t > 0; --t) {
          if (fs[t] < fs[t - 1]) {
            float a = fs[t]; fs[t] = fs[t - 1]; fs[t - 1] = a;
            int   b = fi[t]; fi[t] = fi[t - 1]; fi[t - 1] = b;
          }
        }
      }
    }
    const int row = xbase + tid;
    float acc = 0.0f;
#pragma unroll
    for (int t = 0; t < KNN; ++t) {
      const float d = sqrtf(fmaxf(fs[t], 1e-12f));
      out_dist[row * KNN + t] = d;
      nbr_idx[row * KNN + t]  = fi[t];
      acc += d;
    }
    dmean[row] = acc * (1.0f / KNN);
  }
}

// ---------------------------------------------------------------------------
// Kernel 3: gradient -> influence = |g| * iw, plus per-row mean / unbiased std.
// g_row = x * sum_j(w_j) - sum_j(m_j * w_j),  w_j = 1/(K * d_j)
// ---------------------------------------------------------------------------
__global__ __launch_bounds__(256) void grad_influence_kernel(
    const float* __restrict__ feat,
    const float* __restrict__ mem,
    const float* __restrict__ iw,
    const float* __restrict__ dist,   // [RROWS][KNN]
    const int*   __restrict__ nbr,    // [RROWS][KNN]
    float* __restrict__ influence,    // [RROWS][DD]
    float* __restrict__ rowstats)     // [RROWS][2]: mean, std
{
  __shared__ float r1[256], r2[256];
  const int row = blockIdx.x;
  const int tid = threadIdx.x;

  float w[KNN]; int id[KNN]; float sw = 0.0f;
#pragma unroll
  for (int j = 0; j < KNN; ++j) {
    const float d = dist[row * KNN + j];
    w[j]  = 1.0f / (KNN * d);
    id[j] = nbr[row * KNN + j];
    sw += w[j];
  }

  const int c0 = tid * 4;  // 256 threads x 4 dims = 1024
  const float4 xv = *reinterpret_cast<const float4*>(&feat[(size_t)row * DD + c0]);
  float g0 = xv.x * sw, g1 = xv.y * sw, g2 = xv.z * sw, g3 = xv.w * sw;
#pragma unroll
  for (int j = 0; j < KNN; ++j) {
    const float4 mv = *reinterpret_cast<const float4*>(&mem[(size_t)id[j] * DD + c0]);
    g0 -= mv.x * w[j]; g1 -= mv.y * w[j]; g2 -= mv.z * w[j]; g3 -= mv.w * w[j];
  }
  const float i0 = fabsf(g0) * iw[c0 + 0];
  const float i1 = fabsf(g1) * iw[c0 + 1];
  const float i2 = fabsf(g2) * iw[c0 + 2];
  const float i3 = fabsf(g3) * iw[c0 + 3];
  float4 outv; outv.x = i0; outv.y = i1; outv.z = i2; outv.w = i3;
  *reinterpret_cast<float4*>(&influence[(size_t)row * DD + c0]) = outv;

  r1[tid] = i0 + i1 + i2 + i3;
  r2[tid] = i0 * i0 + i1 * i1 + i2 * i2 + i3 * i3;
  __syncthreads();
  for (int off = 128; off > 0; off >>= 1) {
    if (tid < off) { r1[tid] += r1[tid + off]; r2[tid] += r2[tid + off]; }
    __syncthreads();
  }
  if (tid == 0) {
    const float s1 = r1[0], s2 = r2[0];
    const float mean = s1 * (1.0f / DD);
    const float var  = fmaxf((s2 - s1 * s1 * (1.0f / DD)) * (1.0f / (DD - 1)), 0.0f);
    rowstats[2 * row + 0] = mean;
    rowstats[2 * row + 1] = sqrtf(var);
  }
}

// ---------------------------------------------------------------------------
// Kernel 4: global mean/std of the distance signal, matching the reference's
// std over the D-broadcast array with ddof=1: var = D*(S2 - S1^2/R)/(R*D - 1)
// ---------------------------------------------------------------------------
__global__ __launch_bounds__(256) void dstats_kernel(const float* __restrict__ dmean,
                                                     float* __restrict__ gstats) {
  __shared__ float r1[256], r2[256];
  float s1 = 0.0f, s2 = 0.0f;
  for (int i = threadIdx.x; i < RROWS; i += 256) {
    const float v = dmean[i];
    s1 += v; s2 += v * v;
  }
  r1[threadIdx.x] = s1;
  r2[threadIdx.x] = s2;
  __syncthreads();
  for (int off = 128; off > 0; off >>= 1) {
    if (threadIdx.x < off) { r1[threadIdx.x] += r1[threadIdx.x + off];
                             r2[threadIdx.x] += r2[threadIdx.x + off]; }
    __syncthreads();
  }
  if (threadIdx.x == 0) {
    const float S1 = r1[0], S2 = r2[0];
    const float mean = S1 / (float)RROWS;
    const float var  = (float)DD * (S2 - S1 * S1 / (float)RROWS) /
                       ((float)RROWS * (float)DD - 1.0f);
    gstats[0] = mean;
    gstats[1] = sqrtf(fmaxf(var, 0.0f));
  }
}

// ---------------------------------------------------------------------------
// Kernel 5: combined -> sigmoid -> noise_std
// ---------------------------------------------------------------------------
__global__ __launch_bounds__(256) void noise_kernel(
    const float* __restrict__ influence,
    const float* __restrict__ rowstats,
    const float* __restrict__ dmean,
    const float* __restrict__ gstats,
    const float* __restrict__ dw,
    float* __restrict__ noise)
{
  const size_t i = (size_t)blockIdx.x * 256 + threadIdx.x;   // RROWS*DD total
  const int row = (int)(i >> 10);
  const float inf = influence[i];
  const float rm = rowstats[2 * row + 0];
  const float rs = rowstats[2 * row + 1];
  const float infn = (inf - rm) / (rs + EPSV);
  const float dn = (dmean[row] - gstats[0]) / (gstats[1] + EPSV);
  const float comb = infn + dw[0] * dn;
  const float sg = 1.0f / (1.0f + expf(-comb));
  noise[i] = 0.01f + 0.49f * sg;
}

// ---------------------------------------------------------------------------
extern "C" void kernel_launch(void* const* d_in, const int* in_sizes, int n_in,
                              void* d_out, int out_size, void* d_ws, size_t ws_size,
                              hipStream_t stream) {
  const float* feat = (const float*)d_in[0];   // [8,1024,1024]
  const float* memb = (const float*)d_in[1];   // [16384,1024]
  const float* iw   = (const float*)d_in[2];   // [1024]
  const float* dw   = (const float*)d_in[3];   // [1]

  float* out = (float*)d_out;
  float* influence = out;                       // 8388608
  float* noise     = out + (size_t)RROWS * DD;  // 8388608
  float* dists     = out + (size_t)2 * RROWS * DD;  // 73728

  float* ws = (float*)d_ws;
  float* x2       = ws;                  // 8192
  float* m2       = ws + 8192;           // 16384
  int*   nbr      = (int*)(ws + 24576);  // 73728
  float* dmean    = ws + 98304;          // 8192
  float* rowstats = ws + 106496;         // 16384
  float* gstats   = ws + 122880;         // 2

  rowsq_kernel<<<RROWS, 256, 0, stream>>>(feat, x2);
  rowsq_kernel<<<MM,    256, 0, stream>>>(memb, m2);

  dist_topk_wmma_kernel<<<RROWS / 16, 256, 0, stream>>>(
      feat, memb, x2, m2, dists, nbr, dmean);

  grad_influence_kernel<<<RROWS, 256, 0, stream>>>(
      feat, memb, iw, dists, nbr, influence, rowstats);

  dstats_kernel<<<1, 256, 0, stream>>>(dmean, gstats);

  noise_kernel<<<(RROWS * DD) / 256, 256, 0, stream>>>(
      influence, rowstats, dmean, gstats, dw, noise);
}